// gsage_layer_13271448945163
// MI455X (gfx1250) — compile-verified
//
#include <hip/hip_runtime.h>

// GraphSAGE 'pool' layer for MI455X (gfx1250, wave32, WMMA).
// Pipeline: [zero h_neigh] -> [pooled = relu(h@Wp+b) via f32 WMMA]
//           -> [edge scatter-max via int-bit atomicMax (pooled>=0)]
//           -> [rst = h@Ws + h_neigh@Wn + bias, fused LayerNorm, via f32 WMMA]
//
// Roofline: GEMMs ~10 GFLOP (trivial) but stream ~51 MB each -> HBM-bound.
// Edge phase: pooled (51 MB) + h_neigh (51 MB) both L2-resident (192 MB L2),
// so the 102M gather+atomic-max ops are served by L2, not HBM.

typedef __attribute__((ext_vector_type(2))) float v2f;
typedef __attribute__((ext_vector_type(8))) float v8f;

#define NF 128
#define LN_EPS 1e-5f

// ---------------------------------------------------------------- zero init
__global__ void __launch_bounds__(256) gsage_zero_kernel(float* p, size_t n) {
    size_t i = (size_t)blockIdx.x * blockDim.x + threadIdx.x;
    const size_t stride = (size_t)gridDim.x * blockDim.x;
    for (; i < n; i += stride) p[i] = 0.0f;
}

// ---------------------------------------------- pooled = relu(h @ Wp + bp)
// Block = 256 threads = 8 wave32s. Wave w owns 16-column strip [16w,16w+16).
// Grid-strides over 16-row tiles; W strip (128x16) hoisted into registers.
__global__ void __launch_bounds__(256) gsage_pool_gemm_kernel(
    const float* __restrict__ h, const float* __restrict__ Wp,
    const float* __restrict__ bp, float* __restrict__ pooled, int n_tiles) {
    const int wave = threadIdx.x >> 5;
    const int lane = threadIdx.x & 31;
    const int hh   = lane >> 4;     // half-wave select
    const int l16  = lane & 15;
    const int col  = (wave << 4) + l16;

    // B strip in registers: breg[kk] = { Wp[k+2h][col], Wp[k+2h+1][col] }, k=4*kk
    v2f breg[32];
#pragma unroll
    for (int kk = 0; kk < 32; ++kk) {
        const int kr = (kk << 2) + (hh << 1);
        breg[kk].x = Wp[kr * NF + col];
        breg[kk].y = Wp[(kr + 1) * NF + col];
    }
    const float bias_v = bp[col];

    for (int rt = blockIdx.x; rt < n_tiles; rt += gridDim.x) {
        const float* arow = h + (size_t)(rt * 16 + l16) * NF;
        v8f c = {};
#pragma unroll
        for (int kk = 0; kk < 32; ++kk) {
            const int k = (kk << 2) + (hh << 1);
            const v2f a = *(const v2f*)(arow + k);  // contiguous pair -> b64 load
            c = __builtin_amdgcn_wmma_f32_16x16x4_f32(
                    false, a, false, breg[kk], (short)0, c, false, false);
        }
        // C/D layout: VGPR r -> row (r + 8*hh), col = col
        const int orow = rt * 16 + (hh << 3);
#pragma unroll
        for (int r = 0; r < 8; ++r) {
            float v = c[r] + bias_v;
            v = fmaxf(v, 0.0f);  // relu -> pooled >= 0 (enables int-bit atomicMax)
            pooled[(size_t)(orow + r) * NF + col] = v;
        }
    }
}

// ------------------------------- h_neigh[dst] = max over edges pooled[src]
// One wave (32 lanes) per edge; each lane handles a 4-feature chunk:
// one b128 gather + 4 non-returning int atomics. pooled >= 0 and init 0 =>
// integer atomicMax on float bits is exact float max, and zero-in-degree
// nodes stay 0 (matches the reference's isfinite fixup).
__global__ void __launch_bounds__(256) gsage_edge_max_kernel(
    const float* __restrict__ pooled, const int* __restrict__ src,
    const int* __restrict__ dst, int* __restrict__ hneigh_i, long long n_work) {
    const long long t = (long long)blockIdx.x * blockDim.x + threadIdx.x;
    if (t >= n_work) return;
    const int e  = (int)(t >> 5);        // edge index (32 lanes per edge)
    const int f4 = (int)(t & 31) << 2;   // this lane's 4-feature chunk
    const int s = src[e];
    const int d = dst[e];
    const float4 v = *(const float4*)(pooled + (size_t)s * NF + f4);  // L2 hit
    int* dp = hneigh_i + (size_t)d * NF + f4;
    atomicMax(dp + 0, __float_as_int(v.x));
    atomicMax(dp + 1, __float_as_int(v.y));
    atomicMax(dp + 2, __float_as_int(v.z));
    atomicMax(dp + 3, __float_as_int(v.w));
}

// ---------- out = LayerNorm(h @ Ws + h_neigh @ Wn + bias) * gamma + beta
// Both GEMMs chain-accumulate into one WMMA C. LN via 8KB LDS tile + wave32
// shfl_xor reductions over 16-lane row groups.
__global__ void __launch_bounds__(256) gsage_out_gemm_ln_kernel(
    const float* __restrict__ h, const float* __restrict__ hneigh,
    const float* __restrict__ Ws, const float* __restrict__ Wn,
    const float* __restrict__ bias, const float* __restrict__ gamma,
    const float* __restrict__ beta, float* __restrict__ out, int n_tiles) {
    __shared__ float tile[16 * NF];

    const int wave = threadIdx.x >> 5;
    const int lane = threadIdx.x & 31;
    const int hh   = lane >> 4;
    const int l16  = lane & 15;
    const int col  = (wave << 4) + l16;

    v2f bs[32], bn[32];
#pragma unroll
    for (int kk = 0; kk < 32; ++kk) {
        const int kr = (kk << 2) + (hh << 1);
        bs[kk].x = Ws[kr * NF + col];
        bs[kk].y = Ws[(kr + 1) * NF + col];
        bn[kk].x = Wn[kr * NF + col];
        bn[kk].y = Wn[(kr + 1) * NF + col];
    }
    const float bias_v = bias[col];
    const int lncol = l16 << 3;  // this lane's 8-col strip for the LN phase

    for (int rt = blockIdx.x; rt < n_tiles; rt += gridDim.x) {
        const size_t rowbase = (size_t)(rt * 16 + l16) * NF;
        const float* arow_s = h + rowbase;
        const float* arow_n = hneigh + rowbase;
        v8f c = {};
#pragma unroll
        for (int kk = 0; kk < 32; ++kk) {
            const int k = (kk << 2) + (hh << 1);
            const v2f a = *(const v2f*)(arow_s + k);
            c = __builtin_amdgcn_wmma_f32_16x16x4_f32(
                    false, a, false, bs[kk], (short)0, c, false, false);
            const v2f b = *(const v2f*)(arow_n + k);
            c = __builtin_amdgcn_wmma_f32_16x16x4_f32(
                    false, b, false, bn[kk], (short)0, c, false, false);
        }
        const int trow = hh << 3;
#pragma unroll
        for (int r = 0; r < 8; ++r)
            tile[(trow + r) * NF + col] = c[r] + bias_v;
        __syncthreads();

        // LayerNorm: wave handles rows {2*wave, 2*wave+1}; 16 lanes x 8 cols each.
        const int lrow = (wave << 1) + hh;
        const float* trp = tile + lrow * NF + lncol;
        float x[8];
        float s = 0.0f, ss = 0.0f;
#pragma unroll
        for (int i = 0; i < 8; ++i) {
            x[i] = trp[i];
            s  += x[i];
            ss += x[i] * x[i];
        }
#pragma unroll
        for (int m = 8; m >= 1; m >>= 1) {  // reduce within 16-lane row group
            s  += __shfl_xor(s, m, 32);
            ss += __shfl_xor(ss, m, 32);
        }
        const float mu   = s * (1.0f / NF);
        const float var  = ss * (1.0f / NF) - mu * mu;
        const float rstd = rsqrtf(var + LN_EPS);
        float* orow = out + (size_t)(rt * 16 + lrow) * NF + lncol;
#pragma unroll
        for (int i = 0; i < 8; ++i)
            orow[i] = (x[i] - mu) * rstd * gamma[lncol + i] + beta[lncol + i];
        __syncthreads();
    }
}

// ---------------------------------------------------------------- launcher
extern "C" void kernel_launch(void* const* d_in, const int* in_sizes, int n_in,
                              void* d_out, int out_size, void* d_ws, size_t ws_size,
                              hipStream_t stream) {
    const float* h     = (const float*)d_in[0];
    const int*   edge  = (const int*)d_in[1];   // [2, E]; row0=src, row1=dst
    const float* Wp    = (const float*)d_in[2];
    const float* bp    = (const float*)d_in[3];
    const float* Ws    = (const float*)d_in[4];
    const float* Wn    = (const float*)d_in[5];
    const float* bias  = (const float*)d_in[6];
    const float* gamma = (const float*)d_in[7];
    const float* beta  = (const float*)d_in[8];
    float* out = (float*)d_out;

    const int n_nodes = in_sizes[0] / NF;        // 100000 (multiple of 16)
    const int n_edges = in_sizes[1] / 2;         // 800000
    const int n_tiles = (n_nodes + 15) / 16;

    // Workspace: pooled [n_tiles*16, 128] then h_neigh [n_tiles*16, 128]
    float* pooled = (float*)d_ws;
    float* hneigh = pooled + (size_t)n_tiles * 16 * NF;

    const int* src = edge;
    const int* dst = edge + n_edges;

    const size_t zn = (size_t)n_tiles * 16 * NF;
    gsage_zero_kernel<<<2048, 256, 0, stream>>>(hneigh, zn);

    gsage_pool_gemm_kernel<<<1024, 256, 0, stream>>>(h, Wp, bp, pooled, n_tiles);

    const long long work = (long long)n_edges * 32;   // 1 wave per edge
    const int eblocks = (int)((work + 255) / 256);
    gsage_edge_max_kernel<<<eblocks, 256, 0, stream>>>(pooled, src, dst,
                                                       (int*)hneigh, work);

    gsage_out_gemm_ln_kernel<<<1024, 256, 0, stream>>>(h, hneigh, Ws, Wn, bias,
                                                       gamma, beta, out, n_tiles);
}